// AdvancedLoss3D_1881195675843
// MI455X (gfx1250) — compile-verified
//
#include <hip/hip_runtime.h>
#include <hip/hip_bf16.h>
#include <math.h>

// Problem constants (match reference: B=4, N=8192, 3-D points)
#define BB 4
#define NN 8192
#define NT (NN / 16)     // 512 column tiles of 16
#define WAVES 8          // waves per block (wave32 -> 256 threads)
#define RT 2             // row tiles per wave (each B fragment feeds RT WMMAs)

typedef __attribute__((ext_vector_type(2))) float v2f;  // A/B fragment: 16x4 f32 = 2 VGPRs/lane
typedef __attribute__((ext_vector_type(8))) float v8f;  // C/D fragment: 16x16 f32 = 8 VGPRs/lane

// Workspace layout (in floats).
// Augmented-coordinate trick: with A-row a' = (-2x0,-2x1,-2x2, 1) and
// B-col b' = (y0,y1,y2,|y|^2), the K=4 WMMA dot gives  -2 x.y + |y|^2,
// so d^2 = wmma_out + |x|^2  (one add per element, exact f32).
#define OFF_XA   0                        // [B][N][4] pred, A-form (-2p, 1)
#define OFF_YA   (OFF_XA + BB*NN*4)       // [B][N][4] targ, A-form (-2t, 1)
#define OFF_XB   (OFF_YA + BB*NN*4)       // [B][N][4] pred, B-form (p, |p|^2)
#define OFF_YB   (OFF_XB + BB*NN*4)       // [B][N][4] targ, B-form (t, |t|^2)
#define OFF_X2   (OFF_YB + BB*NN*4)       // [B][N] |p|^2
#define OFF_Y2   (OFF_X2 + BB*NN)         // [B][N] |t|^2
#define OFF_MINX (OFF_Y2 + BB*NN)         // [B][N] min_m d(p_n, t_m)
#define OFF_MINY (OFF_MINX + BB*NN)       // [B][N] min_n d(t_m, p_n)

// -------------------------------------------------------------------------
// Build augmented A-form / B-form point arrays and squared norms.
// -------------------------------------------------------------------------
__global__ void prep_kernel(const float* __restrict__ pred,
                            const float* __restrict__ targ,
                            float* __restrict__ ws) {
  int tid = blockIdx.x * blockDim.x + threadIdx.x;
  if (tid >= BB * NN) return;
  float px = pred[tid * 3 + 0], py = pred[tid * 3 + 1], pz = pred[tid * 3 + 2];
  float tx = targ[tid * 3 + 0], ty = targ[tid * 3 + 1], tz = targ[tid * 3 + 2];
  float p2 = px * px + py * py + pz * pz;
  float t2 = tx * tx + ty * ty + tz * tz;
  ((float4*)(ws + OFF_XA))[tid] = make_float4(-2.f * px, -2.f * py, -2.f * pz, 1.f);
  ((float4*)(ws + OFF_YA))[tid] = make_float4(-2.f * tx, -2.f * ty, -2.f * tz, 1.f);
  ((float4*)(ws + OFF_XB))[tid] = make_float4(px, py, pz, p2);
  ((float4*)(ws + OFF_YB))[tid] = make_float4(tx, ty, tz, t2);
  ws[OFF_X2 + tid] = p2;
  ws[OFF_Y2 + tid] = t2;
}

// -------------------------------------------------------------------------
// For each point p_i of P (A-form), min over all q_j of Q (B-form) of
// ||p_i - q_j||.  One wave owns RT=2 adjacent 16-row tiles so every B
// fragment load feeds two v_wmma_f32_16x16x4_f32.  We track min-d^2 and
// defer clamp+sqrt to the very end (both are monotone).
//
// Fragment layout (ISA 7.12.2, 32-bit A 16x4): lane l (l15=l&15, half=l>>4)
//   A: v0 = A[m=l15][k=2*half], v1 = A[m=l15][k=2*half+1]   (B mirrored)
//   D: VGPR v, lane l -> m = v + 8*half, n = l15
// -------------------------------------------------------------------------
__global__ __launch_bounds__(WAVES * 32)
void chamfer_min_kernel(const float* __restrict__ A4,   // [B][N][4] A-form rows
                        const float* __restrict__ Pn2,  // [B][N] row |p|^2
                        const float* __restrict__ B4,   // [B][N][4] B-form cols
                        float* __restrict__ minOut) {   // [B][N]
  const int lane = threadIdx.x & 31;
  const int wave = threadIdx.x >> 5;
  const int b    = blockIdx.y;
  const int rbase0 = (blockIdx.x * WAVES + wave) * (16 * RT);
  const int rbase1 = rbase0 + 16;
  const int half = lane >> 4;
  const int l15  = lane & 15;

  const float* A4b  = A4  + (size_t)b * NN * 4;
  const float* B4b  = B4  + (size_t)b * NN * 4;
  const float* Pn2b = Pn2 + (size_t)b * NN;

  // Loop-invariant A fragments and row norms for both row tiles.
  v2f afrag0 = *(const v2f*)(A4b + (size_t)(rbase0 + l15) * 4 + half * 2);
  v2f afrag1 = *(const v2f*)(A4b + (size_t)(rbase1 + l15) * 4 + half * 2);
  float x2v0[8], x2v1[8];
#pragma unroll
  for (int v = 0; v < 8; ++v) {
    x2v0[v] = Pn2b[rbase0 + v + half * 8];
    x2v1[v] = Pn2b[rbase1 + v + half * 8];
  }

  float mn0[8], mn1[8];
#pragma unroll
  for (int v = 0; v < 8; ++v) { mn0[v] = 3.0e38f; mn1[v] = 3.0e38f; }

#pragma unroll 2
  for (int ct = 0; ct < NT; ++ct) {
    v2f bfrag = *(const v2f*)(B4b + (size_t)(ct * 16 + l15) * 4 + half * 2);
    v8f c0 = {};
    v8f c1 = {};
    // 8 args: (neg_a, A, neg_b, B, c_mod, C, reuse_a, reuse_b)
    c0 = __builtin_amdgcn_wmma_f32_16x16x4_f32(false, afrag0, false, bfrag,
                                               (short)0, c0, false, false);
    c1 = __builtin_amdgcn_wmma_f32_16x16x4_f32(false, afrag1, false, bfrag,
                                               (short)0, c1, false, false);
#pragma unroll
    for (int v = 0; v < 8; ++v) {
      mn0[v] = fminf(mn0[v], c0[v] + x2v0[v]);  // d^2 = (-2x.y + |y|^2) + |x|^2
      mn1[v] = fminf(mn1[v], c1[v] + x2v1[v]);
    }
  }

  // Min across the 16 lanes of each half, then clamp+sqrt once.
#pragma unroll
  for (int v = 0; v < 8; ++v) {
    float m0 = mn0[v], m1 = mn1[v];
    m0 = fminf(m0, __shfl_xor(m0, 1, 32));
    m1 = fminf(m1, __shfl_xor(m1, 1, 32));
    m0 = fminf(m0, __shfl_xor(m0, 2, 32));
    m1 = fminf(m1, __shfl_xor(m1, 2, 32));
    m0 = fminf(m0, __shfl_xor(m0, 4, 32));
    m1 = fminf(m1, __shfl_xor(m1, 4, 32));
    m0 = fminf(m0, __shfl_xor(m0, 8, 32));
    m1 = fminf(m1, __shfl_xor(m1, 8, 32));
    mn0[v] = m0; mn1[v] = m1;
  }
  if (l15 == 0) {
#pragma unroll
    for (int v = 0; v < 8; ++v) {
      minOut[(size_t)b * NN + rbase0 + v + half * 8] = sqrtf(fmaxf(mn0[v], 0.f));
      minOut[(size_t)b * NN + rbase1 + v + half * 8] = sqrtf(fmaxf(mn1[v], 0.f));
    }
  }
}

// -------------------------------------------------------------------------
// Single-block final reduction: vertex MSE, smoothness, symmetry, chamfer.
// -------------------------------------------------------------------------
__global__ __launch_bounds__(256)
void finalize_kernel(const float* __restrict__ pred,
                     const float* __restrict__ targ,
                     const float* __restrict__ ws,
                     float* __restrict__ out) {
  const int tid  = threadIdx.x;
  const int nthr = 256;
  float sv = 0.f, ss = 0.f, sy = 0.f, sc = 0.f;

  // vertex_loss: mean((pred-targ)^2) over B*N*3
  for (int i = tid; i < BB * NN * 3; i += nthr) {
    float d = pred[i] - targ[i];
    sv += d * d;
  }
  // smoothness: mean over (B, N-1) of ||p[i+1]-p[i]||
  for (int k = tid; k < BB * (NN - 1); k += nthr) {
    int b = k / (NN - 1);
    int i = k - b * (NN - 1);
    const float* p = pred + ((size_t)b * NN + i) * 3;
    float dx = p[3] - p[0], dy = p[4] - p[1], dz = p[5] - p[2];
    ss += sqrtf(dx * dx + dy * dy + dz * dz);
  }
  // symmetry: left = p[i], right = p[N-1-i] with x negated; MSE over (B,mid,3)
  const int mid = NN / 2;
  for (int k = tid; k < BB * mid; k += nthr) {
    int b = k / mid;
    int i = k - b * mid;
    const float* l = pred + ((size_t)b * NN + i) * 3;
    const float* r = pred + ((size_t)b * NN + (NN - 1 - i)) * 3;
    float d0 = l[0] + r[0];   // l0 - (-r0)
    float d1 = l[1] - r[1];
    float d2 = l[2] - r[2];
    sy += d0 * d0 + d1 * d1 + d2 * d2;
  }
  // chamfer: sum of both directional mins
  const float* minX = ws + OFF_MINX;
  const float* minY = ws + OFF_MINY;
  for (int i = tid; i < BB * NN; i += nthr) sc += minX[i] + minY[i];

  __shared__ float red[4][256];
  red[0][tid] = sv; red[1][tid] = ss; red[2][tid] = sy; red[3][tid] = sc;
  __syncthreads();
  for (int s = nthr / 2; s > 0; s >>= 1) {
    if (tid < s) {
      red[0][tid] += red[0][tid + s];
      red[1][tid] += red[1][tid + s];
      red[2][tid] += red[2][tid + s];
      red[3][tid] += red[3][tid + s];
    }
    __syncthreads();
  }
  if (tid == 0) {
    float vertex = red[0][0] / (float)(BB * NN * 3);
    float smooth = red[1][0] / (float)(BB * (NN - 1));
    float sym    = red[2][0] / (float)(BB * mid * 3);
    float cham   = red[3][0] / (float)(BB * NN);  // both terms share 1/(B*N)
    out[0] = 1.0f * vertex + 0.1f * smooth + 0.05f * sym + 0.1f * cham;
  }
}

extern "C" void kernel_launch(void* const* d_in, const int* in_sizes, int n_in,
                              void* d_out, int out_size, void* d_ws, size_t ws_size,
                              hipStream_t stream) {
  const float* pred = (const float*)d_in[0];   // (B, N*3) f32
  const float* targ = (const float*)d_in[1];   // (B, N*3) f32
  float* ws  = (float*)d_ws;
  float* out = (float*)d_out;

  prep_kernel<<<(BB * NN + 255) / 256, 256, 0, stream>>>(pred, targ, ws);

  dim3 grid(NN / 16 / (WAVES * RT), BB);   // (32, 4), 256 threads = 8 waves
  // Pass 1: for each pred point, min distance to target set.
  chamfer_min_kernel<<<grid, WAVES * 32, 0, stream>>>(
      ws + OFF_XA, ws + OFF_X2, ws + OFF_YB, ws + OFF_MINX);
  // Pass 2: roles swapped (distance is symmetric) -> no atomics needed.
  chamfer_min_kernel<<<grid, WAVES * 32, 0, stream>>>(
      ws + OFF_YA, ws + OFF_Y2, ws + OFF_XB, ws + OFF_MINY);

  finalize_kernel<<<1, 256, 0, stream>>>(pred, targ, ws, out);
}